// DeepSigmoidFlow_47218870452709
// MI455X (gfx1250) — compile-verified
//
#include <hip/hip_runtime.h>

typedef float v2f __attribute__((ext_vector_type(2)));
typedef float v8f __attribute__((ext_vector_type(8)));

#define EPSf      1e-6f
#define INVf      0.5413233f          /* log(exp(1-EPS)-1) */
#define LOG100f   4.6051701860f
#define LOG1mEPSf (-1.0000005e-6f)    /* log(1-EPS) */

// ws layout (floats)
#define OFF_E   0        /* 3 x 48x48 = 6912 : exp(w_mid[i]) */
#define OFF_EU  6912     /* 48x48           : exp(u_last)    */
#define OFF_EW  9216     /* 48              : exp(w_last)    */
#define OFF_XS1 9264     /* 64  : sum(x after layer0) per b  */
#define OFF_XS2 9328     /* 64  : sum(x after layer1) per b  */
#define OFF_X2  9392     /* 64x48 : x after layer2 per b     */

__device__ __forceinline__ float softplusf(float x) {
    return fmaxf(x, 0.f) + log1pf(expf(-fabsf(x)));
}
__device__ __forceinline__ float sigmf(float x) { return 1.f / (1.f + expf(-x)); }
__device__ __forceinline__ float logcf(float x) { return logf(x * 100.f) - LOG100f; }

__device__ __forceinline__ v8f wmma_f32(v2f a, v2f b, v8f c) {
    // D(16x16) = A(16x4) * B(4x16) + C, fp32, wave32
    return __builtin_amdgcn_wmma_f32_16x16x4_f32(false, a, false, b, (short)0, c,
                                                 false, false);
}

// ---------------------------------------------------------------- kernel 0
__global__ __launch_bounds__(256) void dsf_prep(const float* __restrict__ w_mid,
                                                const float* __restrict__ u_last,
                                                const float* __restrict__ w_last,
                                                float* __restrict__ ws) {
    int idx = blockIdx.x * 256 + threadIdx.x;
    if (idx < 6912)       ws[OFF_E  + idx]        = expf(w_mid[idx]);
    else if (idx < 9216)  ws[OFF_E  + idx]        = expf(u_last[idx - 6912]);
    else if (idx < 9264)  ws[OFF_EW + idx - 9216] = expf(w_last[idx - 9216]);
}

// ---------------------------------------------------------------- kernel 1
// x chain: only s==0 matters for x.  One block per b.
__global__ __launch_bounds__(64) void dsf_phase1(const float* __restrict__ P,
                                                 const float* __restrict__ x0,
                                                 float* __restrict__ ws,
                                                 float* __restrict__ out) {
    const int b = blockIdx.x, tid = threadIdx.x;
    __shared__ float she[48], shq[48], shx[48], shed[48], shedx[48];
    __shared__ float shsc;
    const float* base = P + (size_t)b * 196608;  // (b,0,:)
    float Xin = x0[b];

    for (int i = 0; i < 3; ++i) {
        const float* p = base + i * 192;
        if (tid < 48) {
            float av = softplusf(p[tid] + INVf) + EPSf;
            float sg = sigmf(av * Xin + p[48 + tid]);
            float ev = expf(p[96 + tid]);
            she[tid] = ev; shq[tid] = ev * sg;
        }
        __syncthreads();
        if (tid < 48) {
            const float* E = ws + OFF_E + i * 2304 + tid * 48;
            float den = 0.f, num = 0.f;
            for (int k = 0; k < 48; ++k) { den += E[k] * she[k]; num += E[k] * shq[k]; }
            float xc = (num / den) * (1.f - EPSf) + 0.5f * EPSf;
            shx[tid] = logcf(xc) - logcf(1.f - xc);
        }
        __syncthreads();
        if (tid == 0) {
            float s = 0.f;
            for (int j = 0; j < 48; ++j) s += shx[j];
            shsc = s;
            if (i == 0) ws[OFF_XS1 + b] = s;
            else if (i == 1) ws[OFF_XS2 + b] = s;
        }
        __syncthreads();
        Xin = shsc;
        if (i == 2 && tid < 48) ws[OFF_X2 + b * 48 + tid] = shx[tid];
        __syncthreads();
    }
    // ---- layer 3 (x output is scalar)
    const float* p = base + 576;
    if (tid < 48) {
        float ed = expf(p[144 + tid]);
        shed[tid] = ed; shedx[tid] = ed * shx[tid];
    }
    __syncthreads();
    if (tid < 48) {
        float av = softplusf(p[tid] + INVf) + EPSf;
        const float* EU = ws + OFF_EU + tid * 48;
        float dU = 0.f, tt = 0.f;
        for (int k = 0; k < 48; ++k) { dU += EU[k] * shed[k]; tt += EU[k] * shedx[k]; }
        float sg = sigmf(av * tt / dU + p[48 + tid]);
        float ev = expf(p[96 + tid]);
        she[tid] = ev; shq[tid] = ev * sg;
    }
    __syncthreads();
    if (tid == 0) {
        float dw = 0.f, nw = 0.f;
        for (int k = 0; k < 48; ++k) { dw += ws[OFF_EW + k] * she[k]; nw += ws[OFF_EW + k] * shq[k]; }
        float xc = (nw / dw) * (1.f - EPSf) + 0.5f * EPSf;
        out[b] = logcf(xc) - logcf(1.f - xc);
    }
}

// ---------------------------------------------------------------- kernel 2
// logdet for all 16384 (b,s) sites.  64 threads = 2 waves; 16 sites per wave.
__global__ __launch_bounds__(64) void dsf_main(const float* __restrict__ P,
                                               const float* __restrict__ x0,
                                               const float* __restrict__ ws,
                                               float* __restrict__ out) {
    __shared__ float matbuf[2304];
    __shared__ float ewS[48];
    __shared__ float x2S[48];
    __shared__ float warr[2 * 5 * 768];  // per-wave: a, sg/bb, e, ed/vC, ld2

    const int tid  = threadIdx.x;
    const int lane = tid & 31;
    const int wv   = tid >> 5;
    const int ln15 = lane & 15;
    const int hf   = lane >> 4;
    const int blockSiteBase = blockIdx.x * 32;
    const int b        = blockSiteBase >> 8;        // 32 | 256 -> one b per block
    const int siteBase = blockSiteBase + wv * 16;

    float* aS  = warr + wv * 3840;
    float* sgS = aS + 768;    // doubles as bbS on the last layer
    float* eS  = aS + 1536;
    float* edS = aS + 2304;   // doubles as vC buffer
    float* ldS = aS + 3072;

    if (tid < 48) { ewS[tid] = ws[OFF_EW + tid]; x2S[tid] = ws[OFF_X2 + b * 48 + tid]; }
    const float xin0 = x0[b];
    const float xs1  = ws[OFF_XS1 + b];
    const float xs2  = ws[OFF_XS2 + b];

    float acc[8];
#pragma unroll
    for (int r = 0; r < 8; ++r) acc[r] = 0.f;

    // ======================= middle layers 0..2 =======================
    for (int i = 0; i < 3; ++i) {
        for (int t = tid; t < 2304; t += 64) matbuf[t] = ws[OFF_E + i * 2304 + t];
        __syncthreads();
        const float Xin = (i == 0) ? xin0 : (i == 1 ? xs1 : xs2);

        for (int st = 0; st < 16; ++st) {
            const float* p = P + (size_t)(siteBase + st) * 768 + i * 192;
            for (int j = lane; j < 48; j += 32) {
                float av = softplusf(p[j] + INVf) + EPSf;
                float sg = sigmf(av * Xin + p[48 + j]);
                float ev = expf(p[96 + j]);
                aS[st * 48 + j] = av; sgS[st * 48 + j] = sg; eS[st * 48 + j] = ev;
            }
        }
        __syncthreads();

        v8f den[3], num[3], mm[3];
        for (int jt = 0; jt < 3; ++jt) {
            v8f c0 = {0.f,0.f,0.f,0.f,0.f,0.f,0.f,0.f};
            v8f c1 = c0, c2 = c0;
            const int jc = jt * 16 + ln15;
            for (int kc = 0; kc < 12; ++kc) {
                const int kA = kc * 4 + hf * 2;   // A: lanes0-15 K{0,1}, lanes16-31 K{2,3}
                const int kB = kc * 4 + hf;       // B: rows kB and kB+2
                float e0 = eS[ln15 * 48 + kA],  e1 = eS[ln15 * 48 + kA + 1];
                float s0 = sgS[ln15 * 48 + kA], s1 = sgS[ln15 * 48 + kA + 1];
                float a0 = aS[ln15 * 48 + kA],  a1 = aS[ln15 * 48 + kA + 1];
                v2f bf; bf.x = matbuf[jc * 48 + kB]; bf.y = matbuf[jc * 48 + kB + 2]; // E^T
                v2f af; af.x = e0; af.y = e1;
                c0 = wmma_f32(af, bf, c0);
                af.x = e0 * s0; af.y = e1 * s1;
                c1 = wmma_f32(af, bf, c1);
                af.x = e0 * s0 * (1.f - s0) * a0; af.y = e1 * s1 * (1.f - s1) * a1;
                c2 = wmma_f32(af, bf, c2);
            }
            den[jt] = c0; num[jt] = c1; mm[jt] = c2;
        }

        float ldv[24];
#pragma unroll
        for (int jt = 0; jt < 3; ++jt)
#pragma unroll
            for (int r = 0; r < 8; ++r) {
                float dn = den[jt][r], nm = num[jt][r], m = mm[jt][r];
                float xc = (nm / dn) * (1.f - EPSf) + 0.5f * EPSf;
                ldv[jt * 8 + r] = logf(m) - logf(dn) + (2.f * EPSf + LOG1mEPSf)
                                  - logcf(xc) - logcf(1.f - xc);
            }
        if (i < 2) {
#pragma unroll
            for (int r = 0; r < 8; ++r) {
                float pm = fmaxf(ldv[r], fmaxf(ldv[8 + r], ldv[16 + r]));
                for (int m = 1; m < 16; m <<= 1) pm = fmaxf(pm, __shfl_xor(pm, m, 32));
                float se = expf(ldv[r] - pm) + expf(ldv[8 + r] - pm) + expf(ldv[16 + r] - pm);
                for (int m = 1; m < 16; m <<= 1) se += __shfl_xor(se, m, 32);
                acc[r] += pm + logf(se);
            }
        } else {
#pragma unroll
            for (int jt = 0; jt < 3; ++jt)
#pragma unroll
                for (int r = 0; r < 8; ++r)
                    ldS[(r + hf * 8) * 48 + jt * 16 + ln15] = ldv[jt * 8 + r];
        }
        __syncthreads();
    }

    // ======================= last layer =======================
    for (int t = tid; t < 2304; t += 64) matbuf[t] = ws[OFF_EU + t];
    __syncthreads();
    for (int st = 0; st < 16; ++st) {
        const float* p = P + (size_t)(siteBase + st) * 768 + 576;
        for (int j = lane; j < 48; j += 32) {
            float av = softplusf(p[j] + INVf) + EPSf;
            aS[st * 48 + j]  = av;
            sgS[st * 48 + j] = p[48 + j];        // bb
            eS[st * 48 + j]  = expf(p[96 + j]);
            edS[st * 48 + j] = expf(p[144 + j]); // exp(d)
        }
    }
    __syncthreads();

    // denU[site][j] and t[site][j] via WMMA with EU^T
    v8f dU[3], tT[3];
    for (int jt = 0; jt < 3; ++jt) {
        v8f c0 = {0.f,0.f,0.f,0.f,0.f,0.f,0.f,0.f};
        v8f c1 = c0;
        const int jc = jt * 16 + ln15;
        for (int kc = 0; kc < 12; ++kc) {
            const int kA = kc * 4 + hf * 2;
            const int kB = kc * 4 + hf;
            float d0 = edS[ln15 * 48 + kA], d1 = edS[ln15 * 48 + kA + 1];
            v2f bf; bf.x = matbuf[jc * 48 + kB]; bf.y = matbuf[jc * 48 + kB + 2]; // EU^T
            v2f af; af.x = d0; af.y = d1;
            c0 = wmma_f32(af, bf, c0);
            af.x = d0 * x2S[kA]; af.y = d1 * x2S[kA + 1];
            c1 = wmma_f32(af, bf, c1);
        }
        dU[jt] = c0; tT[jt] = c1;
    }
    __syncthreads();

    // elementwise in D-layout: sigm, denW/numW partials, vC -> edS
    float dwp[8], nwp[8];
#pragma unroll
    for (int r = 0; r < 8; ++r) { dwp[r] = 0.f; nwp[r] = 0.f; }
#pragma unroll
    for (int r = 0; r < 8; ++r)
#pragma unroll
        for (int jt = 0; jt < 3; ++jt) {
            const int j = jt * 16 + ln15, st = r + hf * 8;
            float aa = aS[st * 48 + j], bb = sgS[st * 48 + j], ee = eS[st * 48 + j];
            float ps = aa * tT[jt][r] / dU[jt][r] + bb;
            float sg = sigmf(ps);
            float ew = ewS[j];
            dwp[r] += ew * ee;
            nwp[r] += ew * ee * sg;
            edS[st * 48 + j] = ew * ee * sg * (1.f - sg) * aa / dU[jt][r]; // vC (sans 1/denW)
        }
    __syncthreads();

    float Cp[8];
#pragma unroll
    for (int r = 0; r < 8; ++r) {
        float dw = dwp[r], nw = nwp[r];
        for (int m = 1; m < 16; m <<= 1) { dw += __shfl_xor(dw, m, 32); nw += __shfl_xor(nw, m, 32); }
        float xc = (nw / dw) * (1.f - EPSf) + 0.5f * EPSf;
        Cp[r] = acc[r] + LOG1mEPSf + 2.f * EPSf - logcf(xc) - logcf(1.f - xc) - logf(dw);
    }

    // S[site][p] = sum_k vC[site][k] * EU[k][p]  (B = EU, untransposed)
    v8f SS[3];
    for (int jt = 0; jt < 3; ++jt) {
        v8f c0 = {0.f,0.f,0.f,0.f,0.f,0.f,0.f,0.f};
        const int jc = jt * 16 + ln15;
        for (int kc = 0; kc < 12; ++kc) {
            const int kA = kc * 4 + hf * 2;
            const int kB = kc * 4 + hf;
            v2f af; af.x = edS[ln15 * 48 + kA]; af.y = edS[ln15 * 48 + kA + 1];
            v2f bf; bf.x = matbuf[kB * 48 + jc]; bf.y = matbuf[(kB + 2) * 48 + jc];
            c0 = wmma_f32(af, bf, c0);
        }
        SS[jt] = c0;
    }

    // final per-site LSE over p:  out = Cp + M + log( sum_p S[p] * exp(d[p]+ld2[p]-M) )
#pragma unroll
    for (int r = 0; r < 8; ++r) {
        const int st = r + hf * 8;
        const int siteG = siteBase + st;
        const float* pd = P + (size_t)siteG * 768 + 720;  // d = dsp[3][144..191]
        float T0 = pd[ln15]      + ldS[st * 48 + ln15];
        float T1 = pd[16 + ln15] + ldS[st * 48 + 16 + ln15];
        float T2 = pd[32 + ln15] + ldS[st * 48 + 32 + ln15];
        float pm = fmaxf(T0, fmaxf(T1, T2));
        for (int m = 1; m < 16; m <<= 1) pm = fmaxf(pm, __shfl_xor(pm, m, 32));
        float se = SS[0][r] * expf(T0 - pm) + SS[1][r] * expf(T1 - pm) + SS[2][r] * expf(T2 - pm);
        for (int m = 1; m < 16; m <<= 1) se += __shfl_xor(se, m, 32);
        if (ln15 == 0) out[64 + siteG] = Cp[r] + pm + logf(se);
    }
}

// ---------------------------------------------------------------- launch
extern "C" void kernel_launch(void* const* d_in, const int* in_sizes, int n_in,
                              void* d_out, int out_size, void* d_ws, size_t ws_size,
                              hipStream_t stream) {
    const float* P      = (const float*)d_in[0];
    const float* x0     = (const float*)d_in[1];
    /* d_in[2] = u_mid : provably unused (softmax over size-1 axis) */
    const float* w_mid  = (const float*)d_in[3];
    const float* u_last = (const float*)d_in[4];
    const float* w_last = (const float*)d_in[5];
    float* out = (float*)d_out;
    float* ws  = (float*)d_ws;

    dsf_prep  <<<37, 256, 0, stream>>>(w_mid, u_last, w_last, ws);
    dsf_phase1<<<64, 64, 0, stream>>>(P, x0, ws, out);
    dsf_main  <<<512, 64, 0, stream>>>(P, x0, ws, out);
}